// MLPSDF_20349555049036
// MI455X (gfx1250) — compile-verified
//
#include <hip/hip_runtime.h>
#include <math.h>

#define NLVL 16
#define TBLSZ (1u << 19)
#define BLK 256
#define STRIDE 34   // 32 features + pad, even for 8B-aligned v2f LDS access

typedef float v2f __attribute__((ext_vector_type(2)));
typedef float v8f __attribute__((ext_vector_type(8)));

__global__ __launch_bounds__(BLK)
void hashgrid_mlp_kernel(const float* __restrict__ texc,
                         const float* __restrict__ table,
                         const float* __restrict__ W0,
                         const float* __restrict__ W1,
                         const float* __restrict__ W2,
                         float* __restrict__ out,
                         int N)
{
    __shared__ float sh[BLK * STRIDE];
    __shared__ float s_scale[NLVL];
    __shared__ int   s_res[NLVL];   // 0 => hashed level

    const int tid = threadIdx.x;

    // ---- per-level params (double, matches numpy derivation) ----
    if (tid < NLVL) {
        double s = 16.0 * exp2((8.0 * (double)tid) / 15.0) - 1.0;
        long long r = (long long)ceil(s) + 1;
        s_scale[tid] = (float)s;
        s_res[tid]   = (r * r * r <= (long long)TBLSZ) ? (int)r : 0;
    }
    __syncthreads();

    // ---- stage 1: hash-grid encode (one point per thread) ----
    const int i0 = blockIdx.x * BLK + tid;
    const int pi = (i0 < N) ? i0 : (N - 1);     // clamp tail, keep EXEC full
    const float x = texc[(size_t)pi * 3 + 0];
    const float y = texc[(size_t)pi * 3 + 1];
    const float z = texc[(size_t)pi * 3 + 2];

    // unroll x2: keeps ~16 L2 gathers in flight per wave for latency hiding
    #pragma unroll 2
    for (int l = 0; l < NLVL; ++l) {
        const float sc  = s_scale[l];
        const int   res = s_res[l];
        float px = x * sc + 0.5f, py = y * sc + 0.5f, pz = z * sc + 0.5f;
        float gx = floorf(px), gy = floorf(py), gz = floorf(pz);
        float fx = px - gx,    fy = py - gy,    fz = pz - gz;
        float ox = 1.f - fx,   oy = 1.f - fy,   oz = 1.f - fz;
        unsigned ix = (unsigned)gx, iy = (unsigned)gy, iz = (unsigned)gz;

        const v2f* tb = (const v2f*)table + (size_t)l * TBLSZ;

        // compute all 8 indices first so loads issue as one clause
        unsigned idxs[8];
        #pragma unroll
        for (int c = 0; c < 8; ++c) {
            unsigned cx = ix + ((c >> 2) & 1);
            unsigned cy = iy + ((c >> 1) & 1);
            unsigned cz = iz + (c & 1);
            if (res) {
                idxs[c] = cx + cy * (unsigned)res + cz * (unsigned)(res * res);
            } else {
                idxs[c] = (cx ^ (cy * 2654435761u) ^ (cz * 805459861u)) & (TBLSZ - 1u);
            }
        }
        v2f f[8];
        #pragma unroll
        for (int c = 0; c < 8; ++c) f[c] = tb[idxs[c]];

        float a0 = 0.f, a1 = 0.f;
        #pragma unroll
        for (int c = 0; c < 8; ++c) {
            float w = ((c & 4) ? fx : ox) * ((c & 2) ? fy : oy) * ((c & 1) ? fz : oz);
            a0 = fmaf(w, f[c].x, a0);
            a1 = fmaf(w, f[c].y, a1);
        }
        v2f fe = {a0, a1};
        *(v2f*)&sh[tid * STRIDE + 2 * l] = fe;
    }
    __syncthreads();

    // ---- stage 2: MLP layers 1-2 via V_WMMA_F32_16X16X4_F32 ----
    const int lane = tid & 31;
    const int wv   = tid >> 5;
    const int lh   = lane >> 4;   // lane half selects K pair (ISA f32 A/B layout)
    const int ln   = lane & 15;

    // B fragments: B[k][n] = W[n][k]; lane -> n = nt*16+ln, k = 4*kc + 2*lh + {0,1}
    v2f b0[2][8], b1[2][8];
    #pragma unroll
    for (int nt = 0; nt < 2; ++nt) {
        #pragma unroll
        for (int kc = 0; kc < 8; ++kc) {
            const int n   = nt * 16 + ln;
            const int col = kc * 4 + lh * 2;
            b0[nt][kc] = *(const v2f*)(W0 + n * 32 + col);
            b1[nt][kc] = *(const v2f*)(W1 + n * 32 + col);
        }
    }

    #pragma unroll
    for (int tt = 0; tt < 2; ++tt) {
        const int r0 = wv * 32 + tt * 16;   // wave-private 16-row tile

        // layer 1: h1 = relu(h @ W0^T)
        v8f c0 = {0,0,0,0,0,0,0,0};
        v8f c1 = {0,0,0,0,0,0,0,0};
        #pragma unroll
        for (int kc = 0; kc < 8; ++kc) {
            v2f a = *(const v2f*)&sh[(r0 + ln) * STRIDE + kc * 4 + lh * 2];
            c0 = __builtin_amdgcn_wmma_f32_16x16x4_f32(false, a, false, b0[0][kc],
                                                       (short)0, c0, false, false);
            c1 = __builtin_amdgcn_wmma_f32_16x16x4_f32(false, a, false, b0[1][kc],
                                                       (short)0, c1, false, false);
        }
        #pragma unroll
        for (int r = 0; r < 8; ++r) {       // C/D layout: row = r + 8*lh, col = ln
            const int row = r0 + r + lh * 8;
            sh[row * STRIDE + ln]      = fmaxf(c0[r], 0.f);
            sh[row * STRIDE + 16 + ln] = fmaxf(c1[r], 0.f);
        }

        // layer 2: h2 = relu(h1 @ W1^T)
        v8f d0 = {0,0,0,0,0,0,0,0};
        v8f d1 = {0,0,0,0,0,0,0,0};
        #pragma unroll
        for (int kc = 0; kc < 8; ++kc) {
            v2f a = *(const v2f*)&sh[(r0 + ln) * STRIDE + kc * 4 + lh * 2];
            d0 = __builtin_amdgcn_wmma_f32_16x16x4_f32(false, a, false, b1[0][kc],
                                                       (short)0, d0, false, false);
            d1 = __builtin_amdgcn_wmma_f32_16x16x4_f32(false, a, false, b1[1][kc],
                                                       (short)0, d1, false, false);
        }
        #pragma unroll
        for (int r = 0; r < 8; ++r) {
            const int row = r0 + r + lh * 8;
            sh[row * STRIDE + ln]      = fmaxf(d0[r], 0.f);
            sh[row * STRIDE + 16 + ln] = fmaxf(d1[r], 0.f);
        }
    }
    __syncthreads();

    // ---- stage 3: out = h2 @ W2^T  (32 -> 4, scalar FMAs) ----
    if (i0 < N) {
        float o0 = 0.f, o1 = 0.f, o2 = 0.f, o3 = 0.f;
        #pragma unroll
        for (int k = 0; k < 32; ++k) {
            const float h = sh[tid * STRIDE + k];
            o0 = fmaf(h, W2[0 * 32 + k], o0);
            o1 = fmaf(h, W2[1 * 32 + k], o1);
            o2 = fmaf(h, W2[2 * 32 + k], o2);
            o3 = fmaf(h, W2[3 * 32 + k], o3);
        }
        float4 ov = {o0, o1, o2, o3};
        *(float4*)(out + (size_t)i0 * 4) = ov;
    }
}

extern "C" void kernel_launch(void* const* d_in, const int* in_sizes, int n_in,
                              void* d_out, int out_size, void* d_ws, size_t ws_size,
                              hipStream_t stream) {
    const float* texc  = (const float*)d_in[0];
    const float* table = (const float*)d_in[1];
    const float* W0    = (const float*)d_in[2];
    const float* W1    = (const float*)d_in[3];
    const float* W2    = (const float*)d_in[4];
    float* out = (float*)d_out;

    const int N = in_sizes[0] / 3;
    const int blocks = (N + BLK - 1) / BLK;
    hipLaunchKernelGGL(hashgrid_mlp_kernel, dim3(blocks), dim3(BLK), 0, stream,
                       texc, table, W0, W1, W2, out, N);
}